// CapsuleLayer_12884901888220
// MI455X (gfx1250) — compile-verified
//
#include <hip/hip_runtime.h>

// CapsuleLayer dynamic routing, fused single kernel for gfx1250 (MI455X).
// One workgroup per (capsule c, 16-row b-tile): 160 WGs of 256 threads (8 waves).
// priors tiles are recomputed per routing pass via v_wmma_f32_16x16x4_f32 from
// L2-resident x / route_weights; logits are re-derived algebraically
// (l2 = sum_o P*(v0+v1)), so nothing large ever touches LDS or HBM.

typedef float v2f __attribute__((ext_vector_type(2)));
typedef float v8f __attribute__((ext_vector_type(8)));

#define NB    1152
#define CIN   8
#define COUT  16
#define BT    16
#define WAVES 8
#define NPW   (NB / WAVES)   // 144 n-values per wave

__global__ __launch_bounds__(256)
void capsule_route_kernel(const float* __restrict__ x,
                          const float* __restrict__ w,
                          float* __restrict__ out)
{
    __shared__ float red[WAVES][8][32];   // per-wave WMMA accumulators
    __shared__ float denb[WAVES][16];     // per-wave softmax denominators per b
    __shared__ float vArr[2][256];        // squash outputs v0, v1 (16b x 16o)

    const int tid   = threadIdx.x;
    const int lane  = tid & 31;
    const int wv    = tid >> 5;
    const int m     = lane & 15;          // b-row for A / o-col for B
    const int hi    = lane >> 4;
    const int c     = blockIdx.y;
    const int btile = blockIdx.x;
    const int bbase = btile * BT;

    const float* xb = x + (size_t)(bbase + m) * (NB * CIN);     // x[b][n][i]
    const float* wc = w + (size_t)c * (NB * CIN * COUT);        // W[c][n][i][o]
    const int nbase = wv * NPW;

    for (int pass = 0; pass < 3; ++pass) {
        // routing coefficient vector for this pass (per-lane copy of v rows)
        float va[8];
        if (pass == 1) {
            #pragma unroll
            for (int r = 0; r < 8; ++r)
                va[r] = vArr[0][(r + 8 * hi) * 16 + m];
        } else if (pass == 2) {
            #pragma unroll
            for (int r = 0; r < 8; ++r)
                va[r] = vArr[0][(r + 8 * hi) * 16 + m]
                      + vArr[1][(r + 8 * hi) * 16 + m];
        }

        v8f S  = {};
        v8f S2 = {};
        float den[8] = {0.f,0.f,0.f,0.f,0.f,0.f,0.f,0.f};

        for (int j = 0; j < NPW; ++j) {
            const int n = nbase + j;
            // A operand (x tile 16x8): VGPR j holds K = j + 2*hi  -> b64 load
            v2f a0 = *(const v2f*)(xb + n * CIN + 2 * hi);
            v2f a1 = *(const v2f*)(xb + n * CIN + 4 + 2 * hi);
            // B operand (W tile 8x16): VGPR j holds row K = j + 2*hi, col m
            const float* wn = wc + n * (CIN * COUT);
            v2f b0, b1;
            b0[0] = wn[(2 * hi + 0) * COUT + m];
            b0[1] = wn[(2 * hi + 1) * COUT + m];
            b1[0] = wn[(2 * hi + 4) * COUT + m];
            b1[1] = wn[(2 * hi + 5) * COUT + m];

            if (pass == 0) {
                // uniform probs: pure accumulation chain (two independent accums)
                S  = __builtin_amdgcn_wmma_f32_16x16x4_f32(false, a0, false, b0,
                                                           (short)0, S,  false, false);
                S2 = __builtin_amdgcn_wmma_f32_16x16x4_f32(false, a1, false, b1,
                                                           (short)0, S2, false, false);
            } else {
                v8f P = {};
                P = __builtin_amdgcn_wmma_f32_16x16x4_f32(false, a0, false, b0,
                                                          (short)0, P, false, false);
                P = __builtin_amdgcn_wmma_f32_16x16x4_f32(false, a1, false, b1,
                                                          (short)0, P, false, false);
                // logit[n,b] = sum_o P[b,o]*va[b,o]  (butterfly over the 16 o-lanes)
                float t[8];
                #pragma unroll
                for (int r = 0; r < 8; ++r) t[r] = P[r] * va[r];
                #pragma unroll
                for (int msk = 1; msk <= 8; msk <<= 1) {
                    #pragma unroll
                    for (int r = 0; r < 8; ++r)
                        t[r] += __shfl_xor(t[r], msk, 32);
                }
                // online (unnormalized) softmax accumulation
                #pragma unroll
                for (int r = 0; r < 8; ++r) {
                    float wgt = __expf(t[r]);
                    den[r] += wgt;
                    S[r]   += wgt * P[r];
                }
            }
        }
        if (pass == 0) S += S2;

        // cross-wave reduction + squash
        #pragma unroll
        for (int r = 0; r < 8; ++r) red[wv][r][lane] = S[r];
        if (m == 0) {
            #pragma unroll
            for (int r = 0; r < 8; ++r) denb[wv][r + 8 * hi] = den[r];
        }
        __syncthreads();

        {
            const int b  = tid >> 4;
            const int o  = tid & 15;
            const int rr = b & 7;
            const int hh = b >> 3;
            float num = 0.f, dsum = 0.f;
            #pragma unroll
            for (int wq = 0; wq < WAVES; ++wq) {
                num  += red[wq][rr][hh * 16 + o];
                dsum += denb[wq][b];
            }
            if (pass == 0) dsum = (float)NB;    // softmax(0) over n is uniform
            float s  = num / dsum;
            float sq = s * s;
            #pragma unroll
            for (int msk = 1; msk <= 8; msk <<= 1)
                sq += __shfl_xor(sq, msk, 32);  // ||s||^2 over o
            float vv = s * sqrtf(sq) / (1.0f + sq);
            if (pass < 2) vArr[pass][b * 16 + o] = vv;
            else out[((size_t)c * 256 + bbase + b) * 16 + o] = vv;
        }
        __syncthreads();
    }
}

extern "C" void kernel_launch(void* const* d_in, const int* in_sizes, int n_in,
                              void* d_out, int out_size, void* d_ws, size_t ws_size,
                              hipStream_t stream) {
    (void)in_sizes; (void)n_in; (void)out_size; (void)d_ws; (void)ws_size;
    const float* x = (const float*)d_in[0];          // [256,1152,8] f32
    const float* w = (const float*)d_in[1];          // [10,1152,8,16] f32
    float* out = (float*)d_out;                      // [10,256,1,1,16] f32
    dim3 grid(256 / BT, 10, 1);
    dim3 block(256, 1, 1);
    capsule_route_kernel<<<grid, block, 0, stream>>>(x, w, out);
}